// ProgramSynthesisHRM_1958505087654
// MI455X (gfx1250) — compile-verified
//
#include <hip/hip_runtime.h>
#include <hip/hip_bf16.h>

typedef __attribute__((ext_vector_type(2))) float v2f;
typedef __attribute__((ext_vector_type(8))) float v8f;

#define B_  1024
#define S_  128
#define H_  512
#define NP  30     // MAX_NODES == NUM_NODE_TYPES == node_positions
#define HH  256    // H/2

// -------------------------------------------------------------------------
// Kernel A: global_repr = mean over S.  Streams the full 256 MB of
// hidden_states once (the HBM roofline floor, ~11us @ 23.3 TB/s).
// -------------------------------------------------------------------------
__global__ __launch_bounds__(256) void mean_kernel(const float* __restrict__ hs,
                                                   float* __restrict__ grepr) {
    int idx = blockIdx.x * blockDim.x + threadIdx.x;      // over B*H/4
    if (idx >= B_ * H_ / 4) return;
    int b  = idx / (H_ / 4);
    int h4 = idx % (H_ / 4);
    const float4* p = (const float4*)(hs + (size_t)b * S_ * H_) + h4;
    float4 acc = make_float4(0.f, 0.f, 0.f, 0.f);
    for (int s = 0; s < S_; ++s) {
        float4 v = p[s * (H_ / 4)];
        acc.x += v.x; acc.y += v.y; acc.z += v.z; acc.w += v.w;
    }
    const float inv = 1.0f / S_;
    ((float4*)grepr)[idx] = make_float4(acc.x * inv, acc.y * inv, acc.z * inv, acc.w * inv);
}

// -------------------------------------------------------------------------
// Kernel B: num_nodes_logits = global_repr @ w_num^T + b_num
// M=1024 (64 m-tiles, one wave each), N=30 padded to 32, K=512 via
// V_WMMA_F32_16X16X4_F32.  Padded columns: unconditional load from a
// clamped row pointer + lane-level cndmask to zero (no EXEC juggling).
// -------------------------------------------------------------------------
__global__ __launch_bounds__(256) void numnodes_kernel(const float* __restrict__ grepr,
                                                       const float* __restrict__ w_num,
                                                       const float* __restrict__ b_num,
                                                       float* __restrict__ out_num) {
    int wave = (blockIdx.x * blockDim.x + threadIdx.x) >> 5;   // m-tile id, 0..63
    int lane = threadIdx.x & 31;
    int m0    = wave * 16;
    int mrow  = m0 + (lane & 15);
    int khalf = (lane >> 4) * 2;              // lanes 16-31 hold K+2,K+3
    int nlo   = lane & 15;
    int nhi   = 16 + nlo;
    bool hvalid = (nhi < NP);
    const float* wlo = w_num + (size_t)nlo * H_;
    const float* whi = w_num + (size_t)(hvalid ? nhi : 0) * H_;  // clamped, always valid
    const float* ar  = grepr + (size_t)mrow * H_;
    v8f c0 = {}; v8f c1 = {};
    for (int k = 0; k < H_; k += 4) {
        int kx = k + khalf;
        v2f a  = *(const v2f*)(ar + kx);
        v2f b0 = *(const v2f*)(wlo + kx);
        v2f b1 = *(const v2f*)(whi + kx);
        b1.x = hvalid ? b1.x : 0.0f;          // v_cndmask, no branch
        b1.y = hvalid ? b1.y : 0.0f;
        c0 = __builtin_amdgcn_wmma_f32_16x16x4_f32(false, a, false, b0, (short)0, c0, false, false);
        c1 = __builtin_amdgcn_wmma_f32_16x16x4_f32(false, a, false, b1, (short)0, c1, false, false);
    }
    int mbase = m0 + ((lane >> 4) << 3);       // D: vgpr v -> row v (+8 for lanes 16-31)
    for (int v = 0; v < 8; ++v) {
        int m = mbase + v;
        out_num[(size_t)m * NP + nlo] = c0[v] + b_num[nlo];
        if (hvalid) out_num[(size_t)m * NP + nhi] = c1[v] + b_num[nhi];
    }
}

// -------------------------------------------------------------------------
// Kernel C: per-node heads.  Virtual 32-row weight: rows 0-29 = w_type,
// row 30 = w_exist, row 31 = w_val.  One WMMA GEMM, M = B*30 = 30720
// (exactly 1920 m-tiles), N = 32, K = 512.  whi pointer select is
// loop-invariant, so the K-loop is pure load+wmma.
// -------------------------------------------------------------------------
__global__ __launch_bounds__(256) void heads_kernel(const float* __restrict__ hs,
                                                    const float* __restrict__ w_type,
                                                    const float* __restrict__ b_type,
                                                    const float* __restrict__ w_exist,
                                                    const float* __restrict__ b_exist,
                                                    const float* __restrict__ w_val,
                                                    const float* __restrict__ b_val,
                                                    float* __restrict__ out_exist,
                                                    float* __restrict__ out_type,
                                                    float* __restrict__ out_val) {
    int wave = (blockIdx.x * blockDim.x + threadIdx.x) >> 5;   // 0..1919
    int lane = threadIdx.x & 31;
    int m0 = wave * 16;
    int r  = m0 + (lane & 15);                 // row in (B*30)
    int bb = r / NP, pp = r % NP;
    const float* arow = hs + ((size_t)bb * S_ + pp) * H_;      // node p < 30 < S
    int khalf = (lane >> 4) * 2;
    int nlo = lane & 15;
    int nhi = 16 + nlo;
    const float* wlo = w_type + (size_t)nlo * H_;
    const float* whi = (nhi < NP) ? (w_type + (size_t)nhi * H_)
                                  : ((nhi == NP) ? w_exist : w_val);
    v8f c0 = {}; v8f c1 = {};
    for (int k = 0; k < H_; k += 4) {
        int kx = k + khalf;
        v2f a  = *(const v2f*)(arow + kx);
        v2f b0 = *(const v2f*)(wlo + kx);
        v2f b1 = *(const v2f*)(whi + kx);
        c0 = __builtin_amdgcn_wmma_f32_16x16x4_f32(false, a, false, b0, (short)0, c0, false, false);
        c1 = __builtin_amdgcn_wmma_f32_16x16x4_f32(false, a, false, b1, (short)0, c1, false, false);
    }
    int mbase = m0 + ((lane >> 4) << 3);
    for (int v = 0; v < 8; ++v) {
        int m  = mbase + v;
        int mb = m / NP, mp = m % NP;
        out_type[(size_t)mb * NP * NP + mp * NP + nlo] = c0[v] + b_type[nlo];
        if (nhi < NP) {
            out_type[(size_t)mb * NP * NP + mp * NP + nhi] = c1[v] + b_type[nhi];
        } else if (nhi == NP) {
            out_exist[m] = c1[v] + b_exist[0];
        } else {
            out_val[m]   = c1[v] + b_val[0];
        }
    }
}

// -------------------------------------------------------------------------
// Kernel D: adjacency, fused per batch element (one workgroup each).
// Step 1 (WMMA): [ai | bj] = node_hidden(30x512, padded to 32) @ W1comb^T
//   W1comb rows 0-255 = w_adj1[:, :512], rows 256-511 = w_adj1[:, 512:].
//   8 waves x (1 m-tile x 8 n-tiles); A-fragment reused across 8 WMMAs;
//   the 8 B-row base pointers are hoisted out of the K-loop.
// Step 2 (VALU): adj[i,j] = sum_h relu(ai[i,h]+bj[j,h]+b1[h]) * w2[h] + b2,
//   diagonal zeroed.  ai/bj never touch HBM (63 MB intermediate avoided);
//   LDS rows padded to stride 261 for a conflict-free epilogue.
// -------------------------------------------------------------------------
#define AB_STRIDE 261
__global__ __launch_bounds__(256) void adj_kernel(const float* __restrict__ hs,
                                                  const float* __restrict__ w_adj1,
                                                  const float* __restrict__ b_adj1,
                                                  const float* __restrict__ w_adj2,
                                                  const float* __restrict__ b_adj2,
                                                  float* __restrict__ out_adj) {
    __shared__ float s_ab[2 * NP * AB_STRIDE];   // [ai rows | bj rows]
    __shared__ float s_b1[HH];
    __shared__ float s_w2[HH];

    int b    = blockIdx.x;
    int tid  = threadIdx.x;
    int wave = tid >> 5;
    int lane = tid & 31;
    if (tid < HH) { s_b1[tid] = b_adj1[tid]; s_w2[tid] = w_adj2[tid]; }

    int mt      = wave >> 2;                  // 0..1 (rows 0-15 / 16-31; 30,31 pad)
    int ntbase  = (wave & 3) * 8;             // 8 n-tiles per wave -> N=512
    int mrow    = mt * 16 + (lane & 15);
    bool mvalid = (mrow < NP);
    const float* arow = hs + ((size_t)b * S_ + (mvalid ? mrow : 0)) * H_;
    int khalf = (lane >> 4) * 2;

    // loop-invariant B-row pointers (combined W1 row n for this lane, tile t)
    const float* wr[8];
    for (int t = 0; t < 8; ++t) {
        int n = (ntbase + t) * 16 + (lane & 15);          // 0..511
        wr[t] = w_adj1 + (size_t)(n & 255) * (2 * H_) + (n >> 8) * H_;
    }

    v8f acc[8];
    for (int t = 0; t < 8; ++t) acc[t] = (v8f){};

    for (int k = 0; k < H_; k += 4) {
        int kx = k + khalf;
        v2f a = *(const v2f*)(arow + kx);
        a.x = mvalid ? a.x : 0.0f;            // pad rows 30,31 (cndmask, no branch)
        a.y = mvalid ? a.y : 0.0f;
        for (int t = 0; t < 8; ++t) {
            v2f bf = *(const v2f*)(wr[t] + kx);
            acc[t] = __builtin_amdgcn_wmma_f32_16x16x4_f32(false, a, false, bf,
                                                           (short)0, acc[t], false, false);
        }
    }

    // spill D tiles to LDS (ai region then bj region)
    int mbase = mt * 16 + ((lane >> 4) << 3);
    for (int t = 0; t < 8; ++t) {
        int n   = (ntbase + t) * 16 + (lane & 15);
        int reg = n >> 8;                    // 0 = ai, 1 = bj
        int h   = n & 255;
        for (int v = 0; v < 8; ++v) {
            int m = mbase + v;
            if (m < NP) s_ab[(reg * NP + m) * AB_STRIDE + h] = acc[t][v];
        }
    }
    __syncthreads();

    float badj2 = b_adj2[0];
    for (int pair = tid; pair < NP * NP; pair += blockDim.x) {
        int i = pair / NP, j = pair % NP;
        float r = 0.0f;
        if (i != j) {
            const float* ai = s_ab + i * AB_STRIDE;
            const float* bj = s_ab + (NP + j) * AB_STRIDE;
            float a2 = 0.0f;
            for (int h = 0; h < HH; ++h) {
                float v = ai[h] + bj[h] + s_b1[h];
                a2 += fmaxf(v, 0.0f) * s_w2[h];
            }
            r = a2 + badj2;
        }
        out_adj[(size_t)b * NP * NP + pair] = r;
    }
}

// -------------------------------------------------------------------------
extern "C" void kernel_launch(void* const* d_in, const int* in_sizes, int n_in,
                              void* d_out, int out_size, void* d_ws, size_t ws_size,
                              hipStream_t stream) {
    const float* hs      = (const float*)d_in[0];   // (B,S,H)
    const float* w_exist = (const float*)d_in[1];   // (1,H)
    const float* b_exist = (const float*)d_in[2];   // (1,)
    const float* w_type  = (const float*)d_in[3];   // (30,H)
    const float* b_type  = (const float*)d_in[4];   // (30,)
    const float* w_val   = (const float*)d_in[5];   // (1,H)
    const float* b_val   = (const float*)d_in[6];   // (1,)
    const float* w_adj1  = (const float*)d_in[7];   // (256, 1024)
    const float* b_adj1  = (const float*)d_in[8];   // (256,)
    const float* w_adj2  = (const float*)d_in[9];   // (1,256)
    const float* b_adj2  = (const float*)d_in[10];  // (1,)
    const float* w_num   = (const float*)d_in[11];  // (30,H)
    const float* b_num   = (const float*)d_in[12];  // (30,)

    float* out       = (float*)d_out;               // concatenated outputs
    float* out_num   = out;                         // (B,30)      30720
    float* out_exist = out + 30720;                 // (B,30)      30720
    float* out_type  = out + 61440;                 // (B,30,30)  921600
    float* out_val   = out + 983040;                // (B,30)      30720
    float* out_adj   = out + 1013760;               // (B,30,30)  921600

    float* grepr = (float*)d_ws;                    // (B,H) = 2 MB scratch

    mean_kernel<<<(B_ * H_ / 4 + 255) / 256, 256, 0, stream>>>(hs, grepr);
    numnodes_kernel<<<8, 256, 0, stream>>>(grepr, w_num, b_num, out_num);   // 64 waves
    heads_kernel<<<240, 256, 0, stream>>>(hs, w_type, b_type, w_exist, b_exist,
                                          w_val, b_val, out_exist, out_type, out_val);
    adj_kernel<<<B_, 256, 0, stream>>>(hs, w_adj1, b_adj1, w_adj2, b_adj2, out_adj);
}